// MeanAggregator_43327630082573
// MI455X (gfx1250) — compile-verified
//
#include <hip/hip_runtime.h>
#include <hip/hip_bf16.h>

typedef __attribute__((ext_vector_type(2))) float v2f;
typedef __attribute__((ext_vector_type(8))) float v8f;

#define D_IN  128
#define D_OUT 64

// ---------------------------------------------------------------------------
// Kernel 1: zero-init output accumulator and degree array
// ---------------------------------------------------------------------------
__global__ void zero_kernel(float* __restrict__ out, float* __restrict__ deg,
                            int n_out, int n_deg) {
    int idx = blockIdx.x * blockDim.x + threadIdx.x;
    if (idx < n_out) out[idx] = 0.0f;
    if (idx < n_deg) deg[idx] = 0.0f;
}

// ---------------------------------------------------------------------------
// Kernel 2: h = x @ W^T + b  via V_WMMA_F32_16X16X4_F32 (full fp32 precision)
//
// Block = 256 threads = 8 waves. Wave w: rowTile = w&1, nTile = w>>1.
// Block covers 32 rows x 64 cols. Each wave: 16x16 tile, K-loop of 32 WMMAs.
//
// A (16x4 f32) lane layout: lane = {laneM = lane&15, kHalf = lane>>4}
//   VGPR0 = A[M=laneM][K = 2*kHalf],  VGPR1 = A[M=laneM][K = 2*kHalf+1]
// B (4x16 f32) symmetric: VGPR0 = B[K=2*kHalf][N=laneN], VGPR1 = K=2*kHalf+1
//   B[k][n] = W^T[k][n] = W[n][k]  (W is [64,128] row-major)
// C/D (16x16 f32, 8 VGPRs): vgpr r -> M = 8*kHalf + r, N = laneN
// ---------------------------------------------------------------------------
__global__ __launch_bounds__(256) void gemm_wmma_kernel(
    const float* __restrict__ x, const float* __restrict__ W,
    const float* __restrict__ bias, float* __restrict__ h, int n_nodes) {

    const int lane  = threadIdx.x & 31;
    const int wave  = threadIdx.x >> 5;
    const int laneM = lane & 15;          // also laneN for B/C
    const int kHalf = lane >> 4;

    const int rowTile = wave & 1;
    const int nTile   = wave >> 1;
    const int rowBase = blockIdx.x * 32 + rowTile * 16;
    const int col     = nTile * 16 + laneM;   // output column for this lane

    // Clamp A-row so loads stay in bounds and EXEC stays all-ones for WMMA.
    int arow = rowBase + laneM;
    if (arow >= n_nodes) arow = n_nodes - 1;

    const float* xr = x + (size_t)arow * D_IN + 2 * kHalf;   // A base
    const float* wr = W + (size_t)col  * D_IN + 2 * kHalf;   // B base

    v8f acc = {};
#pragma unroll 8
    for (int kk = 0; kk < D_IN; kk += 4) {
        v2f a = *(const v2f*)(xr + kk);
        v2f b = *(const v2f*)(wr + kk);
        acc = __builtin_amdgcn_wmma_f32_16x16x4_f32(
            /*neg_a=*/false, a, /*neg_b=*/false, b,
            /*c_mod=*/(short)0, acc, /*reuse_a=*/false, /*reuse_b=*/false);
    }

    const float bv = bias[col];
    float* hw = h + (size_t)(rowBase + kHalf * 8) * D_OUT + col;

    if (rowBase + 16 <= n_nodes) {
        // Wave-uniform fast path: whole 16-row tile in range -> straight-line
        // stores (16 lanes per half-wave hit 16 consecutive floats per row).
#pragma unroll
        for (int r = 0; r < 8; ++r) {
            hw[(size_t)r * D_OUT] = acc[r] + bv;
        }
    } else {
        // Boundary tile only (last block).
#pragma unroll
        for (int r = 0; r < 8; ++r) {
            int row = rowBase + kHalf * 8 + r;
            if (row < n_nodes) {
                h[(size_t)row * D_OUT + col] = acc[r] + bv;
            }
        }
    }
}

// ---------------------------------------------------------------------------
// Kernel 3: edge aggregation. One wave per edge; 64 features = 2 per lane.
// out[dst][:] += h[src][:]  (hardware f32 atomics, L2-resident);
// deg[dst] += 1 (lane 0).
// ---------------------------------------------------------------------------
__global__ __launch_bounds__(256) void agg_kernel(
    const float* __restrict__ h, const int* __restrict__ row,
    const int* __restrict__ col, float* __restrict__ out,
    float* __restrict__ deg, int n_edges) {

    int gwave = (blockIdx.x * blockDim.x + threadIdx.x) >> 5;
    int lane  = threadIdx.x & 31;
    if (gwave >= n_edges) return;

    int dst = row[gwave];
    int src = col[gwave];

    const float* hs = h   + (size_t)src * D_OUT;
    float*       od = out + (size_t)dst * D_OUT;

    float v0 = hs[lane];
    float v1 = hs[lane + 32];
    atomicAdd(&od[lane],      v0);
    atomicAdd(&od[lane + 32], v1);
    if (lane == 0) atomicAdd(&deg[dst], 1.0f);
}

// ---------------------------------------------------------------------------
// Kernel 4: out[i][f] /= max(deg[i], 1)
// ---------------------------------------------------------------------------
__global__ void norm_kernel(float* __restrict__ out, const float* __restrict__ deg,
                            int total) {
    int idx = blockIdx.x * blockDim.x + threadIdx.x;
    if (idx < total) {
        float d = deg[idx >> 6];   // idx / 64
        out[idx] = out[idx] / fmaxf(d, 1.0f);
    }
}

// ---------------------------------------------------------------------------
extern "C" void kernel_launch(void* const* d_in, const int* in_sizes, int n_in,
                              void* d_out, int out_size, void* d_ws, size_t ws_size,
                              hipStream_t stream) {
    const float* x    = (const float*)d_in[0];   // [N, 128]
    const float* W    = (const float*)d_in[1];   // [64, 128]
    const float* bias = (const float*)d_in[2];   // [64]
    const int*   row  = (const int*)d_in[3];     // [E]
    const int*   col  = (const int*)d_in[4];     // [E]
    float*       out  = (float*)d_out;           // [N, 64]

    const int n_nodes = in_sizes[0] / D_IN;      // 50000
    const int n_edges = in_sizes[3];             // 800000
    const int n_out_el = n_nodes * D_OUT;

    // Workspace layout: h [N*64 f32] | deg [N f32]  (~13 MB)
    float* h   = (float*)d_ws;
    float* deg = h + (size_t)n_nodes * D_OUT;

    // 1) zero accumulators
    zero_kernel<<<(n_out_el + 255) / 256, 256, 0, stream>>>(out, deg, n_out_el, n_nodes);

    // 2) h = x @ W^T + b   (WMMA fp32; 32 rows per block)
    gemm_wmma_kernel<<<(n_nodes + 31) / 32, 256, 0, stream>>>(x, W, bias, h, n_nodes);

    // 3) scatter-add over edges (one wave per edge, 8 edges per block)
    agg_kernel<<<(n_edges + 7) / 8, 256, 0, stream>>>(h, row, col, out, deg, n_edges);

    // 4) mean normalization
    norm_kernel<<<(n_out_el + 255) / 256, 256, 0, stream>>>(out, deg, n_out_el);
}